// DEMONetWeight_3083786518796
// MI455X (gfx1250) — compile-verified
//
#include <hip/hip_runtime.h>
#include <hip/hip_bf16.h>
#include <math.h>

// Problem constants (match reference)
#define NNODES 100000
#define NEDGES 1200000
#define FDIM   64      // IN == H == OUT == 64

typedef __attribute__((ext_vector_type(2))) float v2f;
typedef __attribute__((ext_vector_type(8))) float v8f;

// ---------------------------------------------------------------------------
// Zero-fill kernel (graph-capture-safe; no hipMemset)
// ---------------------------------------------------------------------------
__global__ void fill_zero_kernel(float* __restrict__ p, long long n) {
    long long i = (long long)blockIdx.x * blockDim.x + threadIdx.x;
    if (i < n) p[i] = 0.0f;
}

// ---------------------------------------------------------------------------
// Degree: deg[src[e]] += 1  (native f32 atomic -> global_atomic_add_f32)
// ---------------------------------------------------------------------------
__global__ void degree_kernel(const int* __restrict__ src,
                              float* __restrict__ deg, int e) {
    int i = blockIdx.x * blockDim.x + threadIdx.x;
    if (i < e) unsafeAtomicAdd(&deg[src[i]], 1.0f);
}

// ---------------------------------------------------------------------------
// Neighbor scatter: acc[src[e]][:] += h[dst[e]][:]
// 16 threads per edge, float4 per thread. Node table (25.6MB) is L2-resident,
// so this is L2 atomic-throughput bound; use native f32 atomics.
// ---------------------------------------------------------------------------
__global__ void scatter_add_kernel(const float* __restrict__ h,
                                   const int* __restrict__ src,
                                   const int* __restrict__ dst,
                                   float* __restrict__ acc) {
    long long idx = (long long)blockIdx.x * blockDim.x + threadIdx.x;
    const long long total = (long long)NEDGES * 16;
    if (idx >= total) return;
    int eid = (int)(idx >> 4);
    int fg  = ((int)idx & 15) * 4;
    int s = src[eid];
    int d = dst[eid];
    float4 v = *(const float4*)(h + (size_t)d * FDIM + fg);
    float* p = acc + (size_t)s * FDIM + fg;
    unsafeAtomicAdd(p + 0, v.x);
    unsafeAtomicAdd(p + 1, v.y);
    unsafeAtomicAdd(p + 2, v.z);
    unsafeAtomicAdd(p + 3, v.w);
}

// ---------------------------------------------------------------------------
// Fused layer:  out = elu( h@Wg + (acc*inv_deg)@Wl + (h*deg_pos)@Ws + b )
// One K=192 GEMM over A = [h | acc*inv_deg | h*deg_pos], done as three K=64
// bands with V_WMMA_F32_16X16X4_F32 (full f32 precision, 48 WMMA/tile).
// One wave32 per 16x16 output tile.  EXEC stays all-ones (uniform guards).
//
// Fragment layouts (CDNA5 ISA 7.12.2):
//   A 16x4 : lanes 0-15 -> M=lane, VGPR0=K0,VGPR1=K1 ; lanes16-31 -> K2,K3
//   B 4x16 : VGPR j, lanes 0-15 -> K=j, N=lane ; lanes 16-31 -> K=j+2
//   C/D    : VGPR r, lanes 0-15 -> M=r, N=lane ; lanes 16-31 -> M=r+8
// ---------------------------------------------------------------------------
__global__ __launch_bounds__(256)
void demonet_layer_wmma(const float* __restrict__ hin,
                        const float* __restrict__ acc,
                        const float* __restrict__ deg,
                        const float* __restrict__ Wg,
                        const float* __restrict__ Wl,
                        const float* __restrict__ Ws,
                        const float* __restrict__ bias,
                        float* __restrict__ hout) {
    const int lane = threadIdx.x & 31;
    const int wid  = (int)((blockIdx.x * blockDim.x + threadIdx.x) >> 5);
    const int ntiles = (NNODES / 16) * (FDIM / 16);   // 25000
    if (wid >= ntiles) return;                         // wave-uniform

    const int tile_m = wid >> 2;          // 0..6249
    const int tile_n = (wid & 3) << 4;    // 0,16,32,48

    const int m    = lane & 15;
    const int row  = tile_m * 16 + m;     // A row owned by this lane
    const int col  = tile_n + m;          // B/D column owned by this lane
    const int koff = (lane >> 4) << 1;    // 0 for lanes 0-15, 2 for 16-31

    // Per-row degree terms (each lane owns exactly one A row)
    const float dv  = deg[row];
    const float dp  = dv > 0.0f ? 1.0f : 0.0f;
    const float idg = dp / fmaxf(dv, 1.0f);

    const float cb = bias[col];
    v8f c = {cb, cb, cb, cb, cb, cb, cb, cb};

    const float* __restrict__ arow = hin + (size_t)row * FDIM;
    const float* __restrict__ nrow = acc + (size_t)row * FDIM;

    // Band 1: h @ Wg
#pragma unroll
    for (int k = 0; k < FDIM; k += 4) {
        v2f a = *(const v2f*)(arow + k + koff);
        v2f b;
        b.x = Wg[(k + koff + 0) * FDIM + col];
        b.y = Wg[(k + koff + 1) * FDIM + col];
        c = __builtin_amdgcn_wmma_f32_16x16x4_f32(false, a, false, b,
                                                  (short)0, c, false, false);
    }
    // Band 2: (acc * inv_deg) @ Wl
#pragma unroll
    for (int k = 0; k < FDIM; k += 4) {
        v2f a = *(const v2f*)(nrow + k + koff);
        a.x *= idg; a.y *= idg;
        v2f b;
        b.x = Wl[(k + koff + 0) * FDIM + col];
        b.y = Wl[(k + koff + 1) * FDIM + col];
        c = __builtin_amdgcn_wmma_f32_16x16x4_f32(false, a, false, b,
                                                  (short)0, c, false, false);
    }
    // Band 3: (h * deg_pos) @ Ws
#pragma unroll
    for (int k = 0; k < FDIM; k += 4) {
        v2f a = *(const v2f*)(arow + k + koff);
        a.x *= dp; a.y *= dp;
        v2f b;
        b.x = Ws[(k + koff + 0) * FDIM + col];
        b.y = Ws[(k + koff + 1) * FDIM + col];
        c = __builtin_amdgcn_wmma_f32_16x16x4_f32(false, a, false, b,
                                                  (short)0, c, false, false);
    }

    // ELU + store (D layout: VGPR r -> M = r (+8 for upper half-lanes))
    const int rbase = tile_m * 16 + ((lane >> 4) << 3);
#pragma unroll
    for (int r = 0; r < 8; ++r) {
        float v = c[r];
        v = v > 0.0f ? v : (__expf(v) - 1.0f);
        hout[(size_t)(rbase + r) * FDIM + col] = v;
    }
}

// ---------------------------------------------------------------------------
// Host launcher
// ---------------------------------------------------------------------------
extern "C" void kernel_launch(void* const* d_in, const int* in_sizes, int n_in,
                              void* d_out, int out_size, void* d_ws, size_t ws_size,
                              hipStream_t stream) {
    const float* x   = (const float*)d_in[0];
    const int*   ei  = (const int*)d_in[1];      // (2, E): src row then dst row
    const int*   src = ei;
    const int*   dst = ei + NEDGES;
    const float* Wg1 = (const float*)d_in[2];
    const float* Wl1 = (const float*)d_in[3];
    const float* Ws1 = (const float*)d_in[4];
    const float* b1  = (const float*)d_in[5];
    const float* Wg2 = (const float*)d_in[6];
    const float* Wl2 = (const float*)d_in[7];
    const float* Ws2 = (const float*)d_in[8];
    const float* b2  = (const float*)d_in[9];
    const float* Wgo = (const float*)d_in[10];
    const float* Wlo = (const float*)d_in[11];
    const float* Wso = (const float*)d_in[12];
    const float* bo  = (const float*)d_in[13];
    float* out = (float*)d_out;

    // Workspace layout: deg | acc | h1 | h2  (~77 MB)
    float* deg = (float*)d_ws;
    float* acc = deg + NNODES;
    float* h1  = acc + (size_t)NNODES * FDIM;
    float* h2  = h1  + (size_t)NNODES * FDIM;

    const int TB = 256;
    const long long featElems = (long long)NNODES * FDIM;       // 6.4M
    const int fillFeatBlocks  = (int)((featElems + TB - 1) / TB);
    const int fillDegBlocks   = (NNODES + TB - 1) / TB;
    const int degBlocks       = (NEDGES + TB - 1) / TB;
    const long long scatThreads = (long long)NEDGES * 16;       // 19.2M
    const int scatBlocks      = (int)((scatThreads + TB - 1) / TB);
    const int gemmBlocks      = ((NNODES / 16) * (FDIM / 16) * 32) / TB; // 3125

    // Degrees (shared by all three layers)
    fill_zero_kernel<<<fillDegBlocks, TB, 0, stream>>>(deg, NNODES);
    degree_kernel<<<degBlocks, TB, 0, stream>>>(src, deg, NEDGES);

    // Layer 1: x -> h1
    fill_zero_kernel<<<fillFeatBlocks, TB, 0, stream>>>(acc, featElems);
    scatter_add_kernel<<<scatBlocks, TB, 0, stream>>>(x, src, dst, acc);
    demonet_layer_wmma<<<gemmBlocks, TB, 0, stream>>>(x, acc, deg,
                                                      Wg1, Wl1, Ws1, b1, h1);

    // Layer 2: h1 -> h2
    fill_zero_kernel<<<fillFeatBlocks, TB, 0, stream>>>(acc, featElems);
    scatter_add_kernel<<<scatBlocks, TB, 0, stream>>>(h1, src, dst, acc);
    demonet_layer_wmma<<<gemmBlocks, TB, 0, stream>>>(h1, acc, deg,
                                                      Wg2, Wl2, Ws2, b2, h2);

    // Layer 3: h2 -> out
    fill_zero_kernel<<<fillFeatBlocks, TB, 0, stream>>>(acc, featElems);
    scatter_add_kernel<<<scatBlocks, TB, 0, stream>>>(h2, src, dst, acc);
    demonet_layer_wmma<<<gemmBlocks, TB, 0, stream>>>(h2, acc, deg,
                                                      Wgo, Wlo, Wso, bo, out);
}